// Self_L2_53927609369065
// MI455X (gfx1250) — compile-verified
//
#include <hip/hip_runtime.h>
#include <hip/hip_bf16.h>
#include <math.h>

// Problem constants from the reference.
constexpr int N = 8192;
constexpr int D = 128;

// GEMM tiling: 128x128 C block per 256-thread (8-wave) workgroup.
// Each wave owns a 4x2 arrangement of 16x16 WMMA subtiles.
constexpr int BLK = 128;
constexpr int MT  = 4;   // subtile rows per wave
constexpr int NT  = 2;   // subtile cols per wave

typedef __attribute__((ext_vector_type(2))) float v2f;
typedef __attribute__((ext_vector_type(8))) float v8f;

// Target builtins are only visible in the device compilation pass; checking
// them in the host pass always fails. Guard the sanity check accordingly.
#if defined(__HIP_DEVICE_COMPILE__) && defined(__gfx1250__)
#if !__has_builtin(__builtin_amdgcn_wmma_f32_16x16x4_f32)
#error "wmma f32 16x16x4 builtin not available in device pass"
#endif
#endif

// ---------------------------------------------------------------------------
// Kernel 1: LR[i,d] = log(s[i,d]) - log(t[i,d])
// ---------------------------------------------------------------------------
__global__ __launch_bounds__(256) void log_ratio_kernel(
    const float* __restrict__ s, const float* __restrict__ t,
    float* __restrict__ lr, int n) {
  int i = blockIdx.x * blockDim.x + threadIdx.x;
  int stride = gridDim.x * blockDim.x;
  for (; i < n; i += stride)
    lr[i] = logf(s[i]) - logf(t[i]);
}

// ---------------------------------------------------------------------------
// Kernel 2: per-block partial of sum_{ij} | (LR @ T^T)[i,j] |
// via V_WMMA_F32_16X16X4_F32, abs-reduction fused into the epilogue.
// FUSED=true computes log-ratio on the fly (used only if d_ws is too small).
// ---------------------------------------------------------------------------
template <bool FUSED>
__global__ __launch_bounds__(256) void wmma_gemm_abs(
    const float* __restrict__ lr,    // [N,D] log ratios (if !FUSED)
    const float* __restrict__ src,   // [N,D] (if FUSED)
    const float* __restrict__ tgt,   // [N,D]
    float* __restrict__ partials) {
  const int lane  = threadIdx.x & 31;
  const int wave  = threadIdx.x >> 5;
  const int lm    = lane & 15;        // M (or N) index within subtile
  const int kb    = (lane >> 4) * 2;  // K pair selector within 4-wide k step

  const int waveM = wave >> 2;   // 0..1
  const int waveN = wave & 3;    // 0..3

  const int rowBase = blockIdx.y * BLK + waveM * (MT * 16);
  const int colBase = blockIdx.x * BLK + waveN * (NT * 16);

  v8f acc[MT][NT];
#pragma unroll
  for (int r = 0; r < MT; ++r)
#pragma unroll
    for (int c = 0; c < NT; ++c)
      acc[r][c] = (v8f){0.f, 0.f, 0.f, 0.f, 0.f, 0.f, 0.f, 0.f};

  const float* aRow[MT];
  const float* aRow2[MT];
  const float* bRow[NT];
#pragma unroll
  for (int r = 0; r < MT; ++r) {
    const size_t row = (size_t)(rowBase + r * 16 + lm) * D;
    aRow[r]  = (FUSED ? src : lr) + row;
    aRow2[r] = tgt + row;  // only used when FUSED
  }
#pragma unroll
  for (int c = 0; c < NT; ++c)
    bRow[c] = tgt + (size_t)(colBase + c * 16 + lm) * D;

  for (int k0 = 0; k0 < D; k0 += 4) {
    v2f a[MT], b[NT];
#pragma unroll
    for (int r = 0; r < MT; ++r) {
      if (FUSED) {
        v2f sv = *(const v2f*)(aRow[r]  + k0 + kb);
        v2f tv = *(const v2f*)(aRow2[r] + k0 + kb);
        a[r].x = logf(sv.x) - logf(tv.x);
        a[r].y = logf(sv.y) - logf(tv.y);
      } else {
        a[r] = *(const v2f*)(aRow[r] + k0 + kb);
      }
    }
#pragma unroll
    for (int c = 0; c < NT; ++c)
      b[c] = *(const v2f*)(bRow[c] + k0 + kb);

#pragma unroll
    for (int r = 0; r < MT; ++r)
#pragma unroll
      for (int c = 0; c < NT; ++c)
        acc[r][c] = __builtin_amdgcn_wmma_f32_16x16x4_f32(
            /*neg_a=*/false, a[r], /*neg_b=*/false, b[c],
            /*c_mod=*/(short)0, acc[r][c],
            /*reuse_a=*/false, /*reuse_b=*/false);
  }

  // |-x| == |x|, so the global negation in the reference vanishes here.
  float s = 0.f;
#pragma unroll
  for (int r = 0; r < MT; ++r)
#pragma unroll
    for (int c = 0; c < NT; ++c)
#pragma unroll
      for (int e = 0; e < 8; ++e)
        s += fabsf(acc[r][c][e]);

  // Deterministic wave tree reduction (wave32).
#pragma unroll
  for (int off = 16; off > 0; off >>= 1)
    s += __shfl_xor(s, off, 32);

  __shared__ float wsum[8];
  if (lane == 0) wsum[wave] = s;
  __syncthreads();
  if (threadIdx.x == 0) {
    float tot = 0.f;
#pragma unroll
    for (int w = 0; w < 8; ++w) tot += wsum[w];
    partials[blockIdx.y * gridDim.x + blockIdx.x] = tot;
  }
}

// ---------------------------------------------------------------------------
// Kernel 3: deterministic reduction of the per-block partials + finalize.
// ---------------------------------------------------------------------------
__global__ __launch_bounds__(256) void finalize_kernel(
    const float* __restrict__ partials, int n, float* __restrict__ out) {
  __shared__ float sm[256];
  float s = 0.f;
  for (int i = threadIdx.x; i < n; i += 256) s += partials[i];
  sm[threadIdx.x] = s;
  __syncthreads();
  for (int off = 128; off > 0; off >>= 1) {
    if (threadIdx.x < off) sm[threadIdx.x] += sm[threadIdx.x + off];
    __syncthreads();
  }
  if (threadIdx.x == 0)
    out[0] = 1e-4f + sm[0] / ((float)N * (float)N);
}

// ---------------------------------------------------------------------------
extern "C" void kernel_launch(void* const* d_in, const int* in_sizes, int n_in,
                              void* d_out, int out_size, void* d_ws,
                              size_t ws_size, hipStream_t stream) {
  (void)in_sizes; (void)n_in; (void)out_size;
  const float* s = (const float*)d_in[0];  // new_source [N,D] f32
  const float* t = (const float*)d_in[1];  // new_target [N,D] f32
  float* out = (float*)d_out;

  constexpr int GB = N / BLK;          // 64
  constexpr int NBLK = GB * GB;        // 4096 partials
  const size_t partBytes = (size_t)NBLK * sizeof(float);
  const size_t lrOff = (partBytes + 255) & ~(size_t)255;
  const size_t need = lrOff + (size_t)N * D * sizeof(float);

  float* partials = (float*)d_ws;
  dim3 grid(GB, GB), block(256);

  if (ws_size >= need) {
    float* lr = (float*)((char*)d_ws + lrOff);
    log_ratio_kernel<<<(N * D + 255) / 256, 256, 0, stream>>>(s, t, lr, N * D);
    wmma_gemm_abs<false><<<grid, block, 0, stream>>>(lr, nullptr, t, partials);
  } else {
    wmma_gemm_abs<true><<<grid, block, 0, stream>>>(nullptr, s, t, partials);
  }
  finalize_kernel<<<1, 256, 0, stream>>>(partials, NBLK, out);
}